// UPFDGraphSageNet_24764781429188
// MI455X (gfx1250) — compile-verified
//
#include <hip/hip_runtime.h>

// ---------------------------------------------------------------------------
// UPFD GraphSAGE net on gfx1250 (MI455X).
//  - Edge aggregation: fp32 global atomics (L2-resident feature table)
//  - Node GEMMs: V_WMMA_F32_16X16X4_F32 (exact fp32; GEMM is not the
//    bottleneck -- the scatter is -- so no precision downgrade needed)
//  - Weights staged in LDS in *fragment order*: one ds_load_b128 per
//    (k-step, tile) yields both Wl and Wr WMMA B-fragments, even-aligned.
//  - Fused epilogue: bias + ReLU + residual + LayerNorm per layer
// ---------------------------------------------------------------------------

typedef float v2f __attribute__((ext_vector_type(2)));
typedef float v4f __attribute__((ext_vector_type(4)));
typedef float v8f __attribute__((ext_vector_type(8)));

#define NFEAT 128
#define NGRAPH 512
#define LN_EPS 1e-5f

// ---------------------------------------------------------------- degree cnt
__global__ void count_edges_kernel(const int* __restrict__ dst,
                                   float* __restrict__ cnt, int E) {
    int e = blockIdx.x * blockDim.x + threadIdx.x;
    if (e < E) atomicAdd(&cnt[dst[e]], 1.0f);
}

__global__ void invcnt_kernel(float* __restrict__ cnt, int N) {
    int n = blockIdx.x * blockDim.x + threadIdx.x;
    if (n < N) cnt[n] = 1.0f / fmaxf(cnt[n], 1.0f);
}

// ------------------------------------------------------------- edge scatter
// 32 threads per edge, each handles 4 features (float4 gather + 4 atomics)
__global__ void scatter_kernel(const float* __restrict__ x,
                               const int* __restrict__ src,
                               const int* __restrict__ dst,
                               float* __restrict__ agg, int E) {
    long long id = (long long)blockIdx.x * blockDim.x + threadIdx.x;
    int e = (int)(id >> 5);
    int f = (int)(id & 31) << 2;
    if (e >= E) return;
    int s = src[e], d = dst[e];
    const float4 v = *(const float4*)(x + (long long)s * NFEAT + f);
    float* p = agg + (long long)d * NFEAT + f;
    atomicAdd(p + 0, v.x);
    atomicAdd(p + 1, v.y);
    atomicAdd(p + 2, v.z);
    atomicAdd(p + 3, v.w);
}

// --------------------------------------------------- fused SAGE layer (WMMA)
// out = LN( relu( (agg*inv_deg) @ Wl + b + x @ Wr ) [+ resid] ; gamma, beta )
// 256 threads = 8 wave32; each wave owns a 16-row strip, all 8 N-tiles.
//
// LDS (128 KB dynamic, 32768 floats) holds both weight matrices permuted to
// WMMA fragment order: quad index q = (kk*8 + t)*32 + lane maps to the
// 16-byte group { Wl[k0][n], Wl[k0+1][n], Wr[k0][n], Wr[k0+1][n] } where
// k0 = kk*4 + (lane>>4)*2 and n = t*16 + (lane&15).  After the GEMM, the
// same LDS is reused as per-wave 16x128 staging for the LayerNorm pass.
__global__ __launch_bounds__(256) void sage_wmma_kernel(
    const float* __restrict__ agg, const float* __restrict__ xin,
    const float* __restrict__ resid,   // nullptr for layer 1
    const float* __restrict__ wl, const float* __restrict__ wr,
    const float* __restrict__ bias, const float* __restrict__ gamma,
    const float* __restrict__ beta, const float* __restrict__ invc,
    float* __restrict__ out, int ntiles) {
    extern __shared__ float smem[];          // 32768 floats = 128 KB
    const int tid = threadIdx.x;

    // ---- cooperative fragment-order weight stage
    // source: row-major W[k][n]; 4096 float4 per matrix, 16 per thread
    {
        const float4* a4 = (const float4*)wl;
        const float4* b4 = (const float4*)wr;
#pragma unroll
        for (int i = 0; i < 16; ++i) {
            const int j4 = tid + i * 256;       // float4 index
            const int k = j4 >> 5;              // row (K index)
            const int n0 = (j4 & 31) << 2;      // first column of the quad
            const int kk = k >> 2;
            const int hl = (k >> 1) & 1;        // which half-wave holds k
            const int kj = k & 1;               // element within K pair
            const int t = n0 >> 4;
            const int l0 = n0 & 15;
            // float index of {quad q, slot kj} with q=(kk*8+t)*32+hl*16+l0
            const int base = (((kk * 8 + t) * 32 + hl * 16 + l0) << 2) + kj;
            const float4 a = a4[j4];
            smem[base + 0] = a.x;
            smem[base + 4] = a.y;
            smem[base + 8] = a.z;
            smem[base + 12] = a.w;
            const float4 b = b4[j4];
            smem[base + 2] = b.x;
            smem[base + 6] = b.y;
            smem[base + 10] = b.z;
            smem[base + 14] = b.w;
        }
    }
    __syncthreads();

    const int warp = tid >> 5;
    const int lane = tid & 31;
    const int half = lane >> 4;          // K sub-block select (ISA A/B layout)
    const int l15 = lane & 15;
    const int tile = blockIdx.x * 8 + warp;
    const bool valid = tile < ntiles;

    v8f acc[8];
    if (valid) {
        const int m0 = tile * 16;
        const int rowA = m0 + l15;       // A-matrix: lane -> row
        const float inv = invc[rowA];    // mean-aggregate scale folded in
        const float* aggp = agg + (long long)rowA * NFEAT;
        const float* xp = xin + (long long)rowA * NFEAT;

#pragma unroll
        for (int t = 0; t < 8; ++t) {    // C init = bias broadcast per column
            float bv = bias[t * 16 + l15];
#pragma unroll
            for (int v = 0; v < 8; ++v) acc[t][v] = bv;
        }

        for (int kk = 0; kk < 32; ++kk) {
            const int k0 = kk * 4 + half * 2;   // this half-wave's K pair
            v2f aA, aX;
            aA.x = aggp[k0] * inv;
            aA.y = aggp[k0 + 1] * inv;
            aX.x = xp[k0];
            aX.y = xp[k0 + 1];
#pragma unroll
            for (int t = 0; t < 8; ++t) {
                // one b128: both B-fragments for (kk, t), even-aligned pairs
                const v4f f =
                    *(const v4f*)(smem + (((kk * 8 + t) * 32 + lane) << 2));
                const v2f bL = __builtin_shufflevector(f, f, 0, 1);
                const v2f bR = __builtin_shufflevector(f, f, 2, 3);
                acc[t] = __builtin_amdgcn_wmma_f32_16x16x4_f32(
                    false, aA, false, bL, (short)0, acc[t], false, false);
                acc[t] = __builtin_amdgcn_wmma_f32_16x16x4_f32(
                    false, aX, false, bR, (short)0, acc[t], false, false);
            }
        }
    }
    __syncthreads();   // all waves done reading weights -> LDS reusable

    if (!valid) return;

    // restage ReLU'd 16x128 strip (C/D layout: vgpr->row, lane->col)
    float* stag = smem + warp * 2048;
#pragma unroll
    for (int t = 0; t < 8; ++t)
#pragma unroll
        for (int v = 0; v < 8; ++v) {
            float val = fmaxf(acc[t][v], 0.0f);
            stag[(v + half * 8) * NFEAT + t * 16 + l15] = val;
        }

    const int m0 = tile * 16;
    const int c0 = lane * 4;     // 4 features per lane for the LN pass
    const float4 g4 = *(const float4*)(gamma + c0);
    const float4 be4 = *(const float4*)(beta + c0);

    for (int r = 0; r < 16; ++r) {
        float4 v = *(float4*)(stag + r * NFEAT + c0);
        const long long grow = (long long)(m0 + r);
        if (resid) {
            const float4 h = *(const float4*)(resid + grow * NFEAT + c0);
            v.x += h.x; v.y += h.y; v.z += h.z; v.w += h.w;
        }
        float s = v.x + v.y + v.z + v.w;
        float sq = v.x * v.x + v.y * v.y + v.z * v.z + v.w * v.w;
#pragma unroll
        for (int off = 16; off > 0; off >>= 1) {
            s += __shfl_xor(s, off, 32);
            sq += __shfl_xor(sq, off, 32);
        }
        const float mean = s * (1.0f / 128.0f);
        const float var = sq * (1.0f / 128.0f) - mean * mean;
        const float rstd = rsqrtf(var + LN_EPS);
        float4 o;
        o.x = (v.x - mean) * rstd * g4.x + be4.x;
        o.y = (v.y - mean) * rstd * g4.y + be4.y;
        o.z = (v.z - mean) * rstd * g4.z + be4.z;
        o.w = (v.w - mean) * rstd * g4.w + be4.w;
        *(float4*)(out + grow * NFEAT + c0) = o;
    }
}

// ------------------------------------------------------------- graph pooling
__global__ void pool_scatter_kernel(const float* __restrict__ h3,
                                    const int* __restrict__ batch,
                                    float* __restrict__ gsum,
                                    float* __restrict__ gcnt, int N) {
    long long id = (long long)blockIdx.x * blockDim.x + threadIdx.x;
    int n = (int)(id >> 5);
    int f = (int)(id & 31) << 2;
    if (n >= N) return;
    int b = batch[n];
    const float4 v = *(const float4*)(h3 + (long long)n * NFEAT + f);
    float* p = gsum + (long long)b * NFEAT + f;
    atomicAdd(p + 0, v.x);
    atomicAdd(p + 1, v.y);
    atomicAdd(p + 2, v.z);
    atomicAdd(p + 3, v.w);
    if (f == 0) atomicAdd(&gcnt[b], 1.0f);
}

__global__ void pool_finish_kernel(const float* __restrict__ gsum,
                                   const float* __restrict__ gcnt,
                                   const float* __restrict__ wc,
                                   const float* __restrict__ bc,
                                   float* __restrict__ outc,
                                   float* __restrict__ avg) {
    __shared__ float r0[128];
    __shared__ float r1[128];
    const int g = blockIdx.x, c = threadIdx.x;
    const float inv = 1.0f / fmaxf(gcnt[g], 1.0f);
    const float a = gsum[(long long)g * NFEAT + c] * inv;
    avg[(long long)g * NFEAT + c] = a;
    r0[c] = a * wc[c * 2 + 0];
    r1[c] = a * wc[c * 2 + 1];
    __syncthreads();
    for (int s = 64; s > 0; s >>= 1) {
        if (c < s) { r0[c] += r0[c + s]; r1[c] += r1[c + s]; }
        __syncthreads();
    }
    if (c == 0) {
        outc[g * 2 + 0] = r0[0] + bc[0];
        outc[g * 2 + 1] = r1[0] + bc[1];
    }
}

// ---------------------------------------------------------------------------
extern "C" void kernel_launch(void* const* d_in, const int* in_sizes, int n_in,
                              void* d_out, int out_size, void* d_ws,
                              size_t ws_size, hipStream_t stream) {
    const float* x = (const float*)d_in[0];
    const int* ei = (const int*)d_in[1];
    const int* batch = (const int*)d_in[2];
    const float* w1l = (const float*)d_in[3];
    const float* w1r = (const float*)d_in[4];
    const float* b1 = (const float*)d_in[5];
    const float* g1 = (const float*)d_in[6];
    const float* be1 = (const float*)d_in[7];
    const float* w2l = (const float*)d_in[8];
    const float* w2r = (const float*)d_in[9];
    const float* b2 = (const float*)d_in[10];
    const float* g2 = (const float*)d_in[11];
    const float* be2 = (const float*)d_in[12];
    const float* w3l = (const float*)d_in[13];
    const float* w3r = (const float*)d_in[14];
    const float* b3 = (const float*)d_in[15];
    const float* g3 = (const float*)d_in[16];
    const float* be3 = (const float*)d_in[17];
    const float* wc = (const float*)d_in[18];
    const float* bc = (const float*)d_in[19];

    const int N = in_sizes[0] / NFEAT;   // 50000
    const int E = in_sizes[1] / 2;       // 800000
    const int* src = ei;
    const int* dst = ei + E;

    // ---- workspace layout (floats): agg | h1 | h2 | invcnt | gsum | gcnt
    float* ws = (float*)d_ws;
    const size_t feat = (size_t)N * NFEAT;
    float* agg = ws;
    float* h1 = agg + feat;
    float* h2 = h1 + feat;
    float* inv = h2 + feat;
    float* gsum = inv + N;
    float* gcnt = gsum + (size_t)NGRAPH * NFEAT;

    // ---- output layout: out(512x2) | h3(50000x128) | avg(512x128)
    float* outc = (float*)d_out;
    float* h3 = outc + NGRAPH * 2;
    float* avg = h3 + feat;

    // ---- degree -> 1/max(deg,1)  (edges fixed across layers)
    hipMemsetAsync(inv, 0, (size_t)N * sizeof(float), stream);
    count_edges_kernel<<<(E + 255) / 256, 256, 0, stream>>>(dst, inv, E);
    invcnt_kernel<<<(N + 255) / 256, 256, 0, stream>>>(inv, N);

    const int ntiles = N / 16;               // 3125
    const int sage_blocks = (ntiles + 7) / 8;
    const size_t lds_bytes = 32768 * sizeof(float);   // 128 KB (WGP has 320)
    hipFuncSetAttribute((const void*)sage_wmma_kernel,
                        hipFuncAttributeMaxDynamicSharedMemorySize,
                        (int)lds_bytes);

    const long long sc_threads = (long long)E * 32;
    const int sc_grid = (int)((sc_threads + 255) / 256);
    const size_t aggb = feat * sizeof(float);

    // ---- layer 1 (no residual)
    hipMemsetAsync(agg, 0, aggb, stream);
    scatter_kernel<<<sc_grid, 256, 0, stream>>>(x, src, dst, agg, E);
    sage_wmma_kernel<<<sage_blocks, 256, lds_bytes, stream>>>(
        agg, x, nullptr, w1l, w1r, b1, g1, be1, inv, h1, ntiles);

    // ---- layer 2 (residual h1)
    hipMemsetAsync(agg, 0, aggb, stream);
    scatter_kernel<<<sc_grid, 256, 0, stream>>>(h1, src, dst, agg, E);
    sage_wmma_kernel<<<sage_blocks, 256, lds_bytes, stream>>>(
        agg, h1, h1, w2l, w2r, b2, g2, be2, inv, h2, ntiles);

    // ---- layer 3 (residual h2) -> h3 straight into d_out
    hipMemsetAsync(agg, 0, aggb, stream);
    scatter_kernel<<<sc_grid, 256, 0, stream>>>(h2, src, dst, agg, E);
    sage_wmma_kernel<<<sage_blocks, 256, lds_bytes, stream>>>(
        agg, h2, h2, w3l, w3r, b3, g3, be3, inv, h3, ntiles);

    // ---- global mean pool + classifier
    hipMemsetAsync(gsum, 0, ((size_t)NGRAPH * NFEAT + NGRAPH) * sizeof(float),
                   stream);
    const long long p_threads = (long long)N * 32;
    pool_scatter_kernel<<<(int)((p_threads + 255) / 256), 256, 0, stream>>>(
        h3, batch, gsum, gcnt, N);
    pool_finish_kernel<<<NGRAPH, 128, 0, stream>>>(gsum, gcnt, wc, bc, outc,
                                                   avg);
}